// Drift_62835371540885
// MI455X (gfx1250) — compile-verified
//
#include <hip/hip_runtime.h>
#include <limits>

typedef __attribute__((ext_vector_type(2))) float v2f;
typedef __attribute__((ext_vector_type(4))) float v4f;
typedef __attribute__((ext_vector_type(8))) float v8f;

// ---------------------------------------------------------------------------
// Half-wave exchange: lane i <-> lane i^16 (ds_swizzle group-of-32,
// xor_mask=0x10, or=0, and=0x1f -> imm 0x401F). Wave32.
// ---------------------------------------------------------------------------
__device__ __forceinline__ float xhalf(float v) {
    int i = __builtin_bit_cast(int, v);
    i = __builtin_amdgcn_ds_swizzle(i, 0x401F);
    return __builtin_bit_cast(float, i);
}

__device__ __forceinline__ float selh(int lane, float lo, float hi) {
    return (lane < 16) ? lo : hi;
}

// Single-instruction ReLU: v_med3_f32(x, 0, +inf) == clamp to [0, inf).
__device__ __forceinline__ float relu1(float x) {
    return __builtin_amdgcn_fmed3f(x, 0.0f, __builtin_inff());
}

// D = A(16x4 f32) * B(4x16 f32) + C(16x16 f32), exact fp32 math.
__device__ __forceinline__ v8f wmma4(v2f a, v2f b, v8f c) {
    return __builtin_amdgcn_wmma_f32_16x16x4_f32(
        /*neg_a=*/false, a, /*neg_b=*/false, b,
        /*c_mod=*/(short)0, c, /*reuse_a=*/false, /*reuse_b=*/false);
}

__device__ __forceinline__ void relu8(v8f& h) {
#pragma unroll
    for (int i = 0; i < 8; ++i) h[i] = relu1(h[i]);
}

// ---------------------------------------------------------------------------
// Build one B operand (4x16, K=features 4C..4C+3, N=16 batch rows) from an
// activation tile held in C/D layout:
//   d[i], lanes 0-15  = feature i     (batch row = lane)
//   d[i], lanes 16-31 = feature 8+i   (batch row = lane-16)
// B layout (mirrors the ISA A 16x4 table):
//   b.x lanes0-15 = f(4C),   lanes16-31 = f(4C+2)
//   b.y lanes0-15 = f(4C+1), lanes16-31 = f(4C+3)
// ---------------------------------------------------------------------------
template<int C>
__device__ __forceinline__ v2f bchunk(int lane, const v8f& d) {
    constexpr int r0 = (4 * C) & 7;
    float d0 = d[r0 + 0], d1 = d[r0 + 1], d2 = d[r0 + 2], d3 = d[r0 + 3];
    if constexpr (((4 * C) >> 3) == 0) {
        // source features live in the lower lane half
        v2f b; b.x = selh(lane, d0, xhalf(d2)); b.y = selh(lane, d1, xhalf(d3));
        return b;
    } else {
        // source features live in the upper lane half
        v2f b; b.x = selh(lane, xhalf(d0), d2); b.y = selh(lane, xhalf(d1), d3);
        return b;
    }
}

// ---------------------------------------------------------------------------
// Weight/bias packing into WMMA operand layouts (loaded once per wave).
// ---------------------------------------------------------------------------
__device__ __forceinline__ v2f load_a(const float* __restrict__ W, int out_dim,
                                      int c, int mb, int lane) {
    int m = mb + (lane & 15);
    int k = 4 * c + ((lane < 16) ? 0 : 2);
    bool ok = (m < out_dim);
    v2f a;
    a.x = ok ? W[(size_t)k * out_dim + m]       : 0.0f;
    a.y = ok ? W[(size_t)(k + 1) * out_dim + m] : 0.0f;
    return a;
}

__device__ __forceinline__ v8f load_bias(const float* __restrict__ b, int out_dim,
                                         int mb, int lane) {
    v8f c;
#pragma unroll
    for (int i = 0; i < 8; ++i) {
        int f = mb + i + ((lane < 16) ? 0 : 8);
        c[i] = (f < out_dim) ? b[f] : 0.0f;
    }
    return c;
}

struct Weights {
    v2f a1;         // L1: 4->8,   1 chunk
    v2f a2[2];      // L2: 8->16,  2 chunks
    v2f a3[2][4];   // L3: 16->32, [mtile][chunk]
    v2f a4[8];      // L4: 32->16, 8 chunks
    v2f a5[4];      // L5: 16->8,  4 chunks
    v2f a6[2];      // L6: 8->4,   2 chunks
    v8f c1, c2, c3a, c3b, c4, c5, c6;
};

// ---------------------------------------------------------------------------
// One 16-row tile through the whole net: 25 chained v_wmma_f32_16x16x4_f32.
// Output features 0..3 land in out[0..3], valid in lanes 0-15 (row = lane).
// ---------------------------------------------------------------------------
__device__ __forceinline__ void mlp_tile(const float* __restrict__ x, int row0,
                                         int lane, const Weights& wt, float out[4]) {
    // Input tile as B for layer 1: lane l<16 loads (f0,f1) of row l,
    // lane 16+l loads (f2,f3) of row l — one coalesced b64 per lane.
    const float* p = x + (size_t)(row0 + (lane & 15)) * 4 + ((lane < 16) ? 0 : 2);
    v2f bin = *(const v2f*)p;

    v8f h = wmma4(wt.a1, bin, wt.c1);
    relu8(h);                                            // L1 out (8 feats)

    v8f t = wt.c2;
    t = wmma4(wt.a2[0], bchunk<0>(lane, h), t);
    t = wmma4(wt.a2[1], bchunk<1>(lane, h), t);
    relu8(t);                                            // L2 out (16 feats)

    v8f u0 = wt.c3a, u1 = wt.c3b;
    {
        v2f b0 = bchunk<0>(lane, t), b1 = bchunk<1>(lane, t);
        v2f b2 = bchunk<2>(lane, t), b3 = bchunk<3>(lane, t);
        u0 = wmma4(wt.a3[0][0], b0, u0); u1 = wmma4(wt.a3[1][0], b0, u1);
        u0 = wmma4(wt.a3[0][1], b1, u0); u1 = wmma4(wt.a3[1][1], b1, u1);
        u0 = wmma4(wt.a3[0][2], b2, u0); u1 = wmma4(wt.a3[1][2], b2, u1);
        u0 = wmma4(wt.a3[0][3], b3, u0); u1 = wmma4(wt.a3[1][3], b3, u1);
    }
    relu8(u0); relu8(u1);                                // L3 out (32 feats)

    v8f h4 = wt.c4;
    h4 = wmma4(wt.a4[0], bchunk<0>(lane, u0), h4);
    h4 = wmma4(wt.a4[1], bchunk<1>(lane, u0), h4);
    h4 = wmma4(wt.a4[2], bchunk<2>(lane, u0), h4);
    h4 = wmma4(wt.a4[3], bchunk<3>(lane, u0), h4);
    h4 = wmma4(wt.a4[4], bchunk<0>(lane, u1), h4);
    h4 = wmma4(wt.a4[5], bchunk<1>(lane, u1), h4);
    h4 = wmma4(wt.a4[6], bchunk<2>(lane, u1), h4);
    h4 = wmma4(wt.a4[7], bchunk<3>(lane, u1), h4);
    relu8(h4);                                           // L4 out (16 feats)

    t = wt.c5;
    t = wmma4(wt.a5[0], bchunk<0>(lane, h4), t);
    t = wmma4(wt.a5[1], bchunk<1>(lane, h4), t);
    t = wmma4(wt.a5[2], bchunk<2>(lane, h4), t);
    t = wmma4(wt.a5[3], bchunk<3>(lane, h4), t);
    relu8(t);                                            // L5 out (8 feats)

    v8f o = wt.c6;
    o = wmma4(wt.a6[0], bchunk<0>(lane, t), o);
    o = wmma4(wt.a6[1], bchunk<1>(lane, t), o);

    // Only features 0..3 are real output; ReLU just those.
    out[0] = relu1(o[0]); out[1] = relu1(o[1]);
    out[2] = relu1(o[2]); out[3] = relu1(o[3]);
}

__global__ void __launch_bounds__(256)
mlp_wmma_kernel(const float* __restrict__ x,
                const float* __restrict__ W1, const float* __restrict__ b1,
                const float* __restrict__ W2, const float* __restrict__ b2,
                const float* __restrict__ W3, const float* __restrict__ b3,
                const float* __restrict__ W4, const float* __restrict__ b4,
                const float* __restrict__ W5, const float* __restrict__ b5,
                const float* __restrict__ W6, const float* __restrict__ b6,
                float* __restrict__ y, int npairs) {
    const int lane   = threadIdx.x & 31;
    const int wave   = blockIdx.x * (blockDim.x >> 5) + (threadIdx.x >> 5);
    const int nwaves = gridDim.x * (blockDim.x >> 5);

    // Weights stationary in VGPRs for the whole kernel (one-time load).
    Weights wt;
    wt.a1 = load_a(W1, 8, 0, 0, lane);
    wt.a2[0] = load_a(W2, 16, 0, 0, lane);
    wt.a2[1] = load_a(W2, 16, 1, 0, lane);
#pragma unroll
    for (int c = 0; c < 4; ++c) {
        wt.a3[0][c] = load_a(W3, 32, c, 0,  lane);
        wt.a3[1][c] = load_a(W3, 32, c, 16, lane);
    }
#pragma unroll
    for (int c = 0; c < 8; ++c) wt.a4[c] = load_a(W4, 16, c, 0, lane);
#pragma unroll
    for (int c = 0; c < 4; ++c) wt.a5[c] = load_a(W5, 8, c, 0, lane);
    wt.a6[0] = load_a(W6, 4, 0, 0, lane);
    wt.a6[1] = load_a(W6, 4, 1, 0, lane);
    wt.c1  = load_bias(b1, 8,  0,  lane);
    wt.c2  = load_bias(b2, 16, 0,  lane);
    wt.c3a = load_bias(b3, 32, 0,  lane);
    wt.c3b = load_bias(b3, 32, 16, lane);
    wt.c4  = load_bias(b4, 16, 0,  lane);
    wt.c5  = load_bias(b5, 8,  0,  lane);
    wt.c6  = load_bias(b6, 4,  0,  lane);

    for (int p = wave; p < npairs; p += nwaves) {
        const int row0 = p * 32;

        if (p + nwaves < npairs) {  // pull next iteration's input toward L2/L0
            __builtin_prefetch(x + (size_t)(p + nwaves) * 128 + lane * 4, 0, 3);
        }

        float oa[4], ob[4];
        mlp_tile(x, row0,      lane, wt, oa);   // rows row0   .. row0+15
        mlp_tile(x, row0 + 16, lane, wt, ob);   // rows row0+16.. row0+31

        // Merge: lanes 0-15 store tileA rows, lanes 16-31 store tileB rows.
        v4f o;
#pragma unroll
        for (int j = 0; j < 4; ++j) {
            float bj = xhalf(ob[j]);            // tileB value moved up a half
            o[j] = selh(lane, oa[j], bj);
        }
        __builtin_nontemporal_store(o, (v4f*)(y + (size_t)(row0 + lane) * 4));
    }
}

// ---------------------------------------------------------------------------
// Scalar fallback for rows not covered by full 32-row tiles (N%32 != 0).
// ---------------------------------------------------------------------------
__device__ void dense_relu(const float* in, float* out, const float* __restrict__ W,
                           const float* __restrict__ b, int id, int od) {
    for (int o = 0; o < od; ++o) {
        float s = b[o];
        for (int i = 0; i < id; ++i) s = fmaf(in[i], W[i * od + o], s);
        out[o] = fmaxf(s, 0.0f);
    }
}

__global__ void mlp_tail_kernel(const float* __restrict__ x,
                const float* W1, const float* b1, const float* W2, const float* b2,
                const float* W3, const float* b3, const float* W4, const float* b4,
                const float* W5, const float* b5, const float* W6, const float* b6,
                float* __restrict__ y, int start, int nrows) {
    int r = start + blockIdx.x * blockDim.x + threadIdx.x;
    if (r >= nrows) return;
    float ha[32], hb[32];
    for (int i = 0; i < 4; ++i) ha[i] = x[(size_t)r * 4 + i];
    dense_relu(ha, hb, W1, b1, 4, 8);
    dense_relu(hb, ha, W2, b2, 8, 16);
    dense_relu(ha, hb, W3, b3, 16, 32);
    dense_relu(hb, ha, W4, b4, 32, 16);
    dense_relu(ha, hb, W5, b5, 16, 8);
    dense_relu(hb, ha, W6, b6, 8, 4);
    for (int j = 0; j < 4; ++j) y[(size_t)r * 4 + j] = ha[j];
}

extern "C" void kernel_launch(void* const* d_in, const int* in_sizes, int n_in,
                              void* d_out, int out_size, void* d_ws, size_t ws_size,
                              hipStream_t stream) {
    const float* x  = (const float*)d_in[0];
    const float* W1 = (const float*)d_in[1];  const float* b1 = (const float*)d_in[2];
    const float* W2 = (const float*)d_in[3];  const float* b2 = (const float*)d_in[4];
    const float* W3 = (const float*)d_in[5];  const float* b3 = (const float*)d_in[6];
    const float* W4 = (const float*)d_in[7];  const float* b4 = (const float*)d_in[8];
    const float* W5 = (const float*)d_in[9];  const float* b5 = (const float*)d_in[10];
    const float* W6 = (const float*)d_in[11]; const float* b6 = (const float*)d_in[12];
    float* y = (float*)d_out;

    const int nrows  = in_sizes[0] / 4;
    const int npairs = nrows / 32;   // 32 rows per wave-iteration (two 16-row tiles)

    if (npairs > 0) {
        int blocks = (npairs + 7) / 8;          // 8 waves per 256-thread block
        if (blocks > 1024) blocks = 1024;       // ~8K waves, several pairs each
        mlp_wmma_kernel<<<blocks, 256, 0, stream>>>(
            x, W1, b1, W2, b2, W3, b3, W4, b4, W5, b5, W6, b6, y, npairs);
    }
    const int start = npairs * 32;
    if (start < nrows) {
        int rem = nrows - start;
        mlp_tail_kernel<<<(rem + 255) / 256, 256, 0, stream>>>(
            x, W1, b1, W2, b2, W3, b3, W4, b4, W5, b5, W6, b6, y, start, nrows);
    }
}